// SPNBP_59682865545786
// MI455X (gfx1250) — compile-verified
//
#include <hip/hip_runtime.h>
#include <math.h>

// SPNBP compute_quant_to_alphabet for MI455X (gfx1250, wave32).
//
// out[n, t] = sum_k mp[k,n] * (cdf[k,n,t] - cdf[k,n,t-1]),
//   cdf[k,n,j] = Phi(basec[k,n] + (j-127)*inv[k,n]), j = 0..254,
//   with caps cdf[.,-1] = 0 and cdf[.,255] = 1.
//
// Transcendental-bound. Unique work per pixel n: 64k x 255 edges = 16320 erfc.
// Each edge is shared by two adjacent bins via an LDS table, so the block
// computes each edge exactly once (16 erfc pair-issues/thread instead of 32).
// K-contraction runs through V_WMMA_F32_16X16X4_F32 with a row-replicated
// A operand (A[m,r] = mp[4kk+r] for all m), B[r,c] = probs[4kk+r, bin c];
// 16 chained WMMAs accumulate k = 0..63. Row 0 of D = 16 output bins/wave.

typedef float v2f __attribute__((ext_vector_type(2)));
typedef float v8f __attribute__((ext_vector_type(8)));

#define KCOMP 64     // mixture components
#define NPIX  1024   // N
#define NBINS 256    // 2^b, b=8

__global__ __launch_bounds__(512)
void spnbp_quant_to_alphabet(const float* __restrict__ Q,
                             const float* __restrict__ Q0,
                             const float* __restrict__ base_mean,
                             const float* __restrict__ base_var,
                             const float* __restrict__ mix,
                             float* __restrict__ out) {
    __shared__ float s_mp[KCOMP];
    __shared__ float s_basec[KCOMP];
    __shared__ float s_inv[KCOMP];
    // buf[j][r]: Phi at edge (j-1) for component slot r of the current kk
    // group; buf[0][r] = 0 and buf[256][r] = 1 are the concat caps.
    __shared__ float s_buf[NBINS + 1][4];

    const int n   = blockIdx.x;      // pixel index 0..1023
    const int tid = threadIdx.x;     // 0..511 -> 16 wave32s

    // Stage per-(k,n) scalars into LDS (waves 0-1 fully active: uniform).
    if (tid < KCOMP) {
        const int k      = tid;
        const float qd   = Q[(size_t)n * NPIX + n];     // diag(Q)[n] > 0
        const float q0   = Q0[n];
        const float var  = base_var[k * NPIX + n];
        const float mean = base_mean[k * NPIX + n];
        const float inv  = rsqrtf(qd * qd * var);       // 1/sqrt(qd^2*var)
        s_inv[k]   = inv;
        s_basec[k] = -(qd * mean + q0) * inv;
        s_mp[k]    = mix[k * NPIX + n];                 // mixture_probs[k,n,0]
    }
    if (tid < 4) {                   // caps, written once (wave 0: uniform)
        s_buf[0][tid]     = 0.0f;
        s_buf[NBINS][tid] = 1.0f;
    }
    __syncthreads();

    const int lane = tid & 31;
    const int wave = tid >> 5;                 // 0..15
    const int c    = lane & 15;                // WMMA column = bin within tile
    const int krb  = (lane >= 16) ? 2 : 0;     // K-slot base per ISA f32 layout
    const int t    = wave * 16 + c;            // global bin 0..255

    // Phase-1 work assignment: linear index i = 2*tid covers (j, r) pairs,
    // r = i & 3 (component slot), j = i >> 2 (edge). Each thread does two
    // same-j adjacent-r edges -> packable erfc pair + one ds_store_b64.
    const int e_j  = tid >> 1;                 // edge 0..255 (255 unused)
    const int e_r  = (tid & 1) * 2;            // slot 0 or 2
    const bool e_on = (e_j < NBINS - 1);       // 4*255 = 1020 entries

    const float INV_SQRT2 = 0.70710678118654752f;

    v8f acc = {0.f, 0.f, 0.f, 0.f, 0.f, 0.f, 0.f, 0.f};

    for (int kk = 0; kk < KCOMP / 4; ++kk) {
        // ---- Phase 1: block computes the 4x255 unique CDF edges once. ----
        if (e_on) {
            const int k0 = 4 * kk + e_r;
            const float ej = (float)(e_j - 127);
            const float x0 = s_basec[k0]     + ej * s_inv[k0];
            const float x1 = s_basec[k0 + 1] + ej * s_inv[k0 + 1];
            s_buf[e_j + 1][e_r]     = 0.5f * erfcf(-x0 * INV_SQRT2);
            s_buf[e_j + 1][e_r + 1] = 0.5f * erfcf(-x1 * INV_SQRT2);
        }
        __syncthreads();

        // ---- Phase 2: probs via adjacent-edge differences, then WMMA. ----
        v2f a, b;
        const int k = 4 * kk + krb;
        a[0] = s_mp[k];
        a[1] = s_mp[k + 1];
        b[0] = s_buf[t + 1][krb]     - s_buf[t][krb];      // probs[k,   t]
        b[1] = s_buf[t + 1][krb + 1] - s_buf[t][krb + 1];  // probs[k+1, t]

        // D = A(16x4) x B(4x16) + C ; chained f32 accumulation over k.
        acc = __builtin_amdgcn_wmma_f32_16x16x4_f32(
            /*neg_a=*/false, a, /*neg_b=*/false, b,
            /*c_mod=*/(short)0, acc, /*reuse_a=*/false, /*reuse_b=*/false);

        __syncthreads();   // protect s_buf before next kk overwrites it
    }

    // D VGPR0, lanes 0..15 = row M=0, N=lane -> out for bins [16*wave, +16).
    if (lane < 16) {
        out[(size_t)n * NBINS + t] = acc[0];
    }
}

extern "C" void kernel_launch(void* const* d_in, const int* in_sizes, int n_in,
                              void* d_out, int out_size, void* d_ws, size_t ws_size,
                              hipStream_t stream) {
    (void)in_sizes; (void)n_in; (void)d_ws; (void)ws_size; (void)out_size;
    const float* Q    = (const float*)d_in[0];
    const float* Q0   = (const float*)d_in[1];
    const float* mean = (const float*)d_in[2];
    const float* var  = (const float*)d_in[3];
    const float* mix  = (const float*)d_in[4];
    // d_in[5] is b (=8); NBINS=256 baked into the launch geometry.
    float* out = (float*)d_out;

    spnbp_quant_to_alphabet<<<dim3(NPIX), dim3(512), 0, stream>>>(
        Q, Q0, mean, var, mix, out);
}